// MeanShift_loss_35192962023870
// MI455X (gfx1250) — compile-verified
//
#include <hip/hip_runtime.h>
#include <cstdint>

#define WAVE 32
#define RT 4                    // 16-row tiles per wave (64 rows/wave)

typedef __attribute__((ext_vector_type(2))) float v2f;
typedef __attribute__((ext_vector_type(8))) float v8f;

__device__ __forceinline__ int imin(int a, int b) { return a < b ? a : b; }

// Raw IEEE minimumNumber. Both fminf (llvm.minnum) and
// __builtin_elementwise_minimumnum (llvm.minimumnum) make the AMDGPU backend
// canonicalize (v_max_num v,v) both operands every iteration -- probed in
// rounds 2 and 4. The inline-asm form emits exactly one v_min_num_f32.
// Inputs here are never NaN (finite coordinates).
__device__ __forceinline__ float fmin_raw(float a, float b) {
    float r;
    asm("v_min_num_f32 %0, %1, %2" : "=v"(r) : "v"(a), "v"(b));
    return r;
}

// ---------------------------------------------------------------------------
// Init: fill row-min bit arrays with +inf bit pattern (ws is poisoned 0xAA).
// ---------------------------------------------------------------------------
__global__ void init_bits_kernel(unsigned int* bits, int n) {
    int i = blockIdx.x * blockDim.x + threadIdx.x;
    if (i < n) bits[i] = 0x7F800000u;  // +inf
}

// ---------------------------------------------------------------------------
// Core: for each row m of X (nRows x 3) compute min over all columns n of Y
// (nCols x 3) of squared distance d2 = |x_m|^2 - 2 x_m.y_n + |y_n|^2.
//
// WMMA f32 16x16x4 packing (K padded from 3 to 4):
//   A'[m] = [ax, ay, az, 1]
//   B'[n] = [-2bx, -2by, -2bz, |b_n|^2]
//   => D[m][n] = |b_n|^2 - 2 a_m.b_n          (no per-tile VALU adds)
// The row-constant |a_m|^2 commutes with min over n; added once per strip.
//
// One wave = RT(=4) 16-row tiles: Y load/frag/b2 amortized over 4 WMMAs.
// blockIdx.y picks a chunk of column tiles. The Y tile pointer is wave-
// uniform and software-pipelined one tile ahead; #pragma unroll 2 keeps two
// loads in flight so the backend can use partial s_wait_loadcnt and overlap
// each load with the other half's WMMA+min chain. Partial last tile handled
// in a uniform-branch tail so the main loop is clamp-free.
// ---------------------------------------------------------------------------
__global__ void __launch_bounds__(256)
nn_min_rows_kernel(const float* __restrict__ X, const float* __restrict__ Y,
                   int nRows, int nCols, int tiles, int tilesPerChunk,
                   unsigned int* __restrict__ rowMinBits) {
    const int lane  = threadIdx.x & (WAVE - 1);
    const int strip = blockIdx.x * (blockDim.x >> 5) + (threadIdx.x >> 5);
    const int nStrips = (nRows + 16 * RT - 1) / (16 * RT);
    if (strip >= nStrips) return;              // wave-uniform

    const int hl  = lane >> 4;                 // lane-half: selects K pair
    const int idx = lane & 15;
    const int rowBase = strip * (16 * RT);

    // ---- A fragments (VGPR v holds K = 2*hl + v; same m in both halves) ----
    v2f aF[RT];
    float a2[RT];
#pragma unroll
    for (int rt = 0; rt < RT; ++rt) {
        const int m = imin(rowBase + rt * 16 + idx, nRows - 1);
        const float ax = X[m * 3 + 0], ay = X[m * 3 + 1], az = X[m * 3 + 2];
        a2[rt] = ax * ax + ay * ay + az * az;          // |a|^2 for this row
        aF[rt][0] = hl ? az   : ax;                    // K = 0 or 2
        aF[rt][1] = hl ? 1.0f : ay;                    // K = 1 or 3
    }

    float rmin[RT][8];
#pragma unroll
    for (int rt = 0; rt < RT; ++rt)
#pragma unroll
        for (int j = 0; j < 8; ++j) rmin[rt][j] = __builtin_inff();

    const int mainTiles = nCols >> 4;                 // full 16-col tiles
    const int t0 = blockIdx.y * tilesPerChunk;
    const int t1 = imin(t0 + tilesPerChunk, tiles);
    const int tMainEnd = imin(t1, mainTiles);
    const int laneOff = idx * 3;                      // loop-invariant VGPR off

    // ---- main loop over full tiles: uniform SALU tile pointer, pipelined ---
    if (t0 < tMainEnd) {
        float bx, by, bz;
        {
            const float* __restrict__ yp = Y + (size_t)t0 * 48;  // 16 pts * 3
            bx = yp[laneOff + 0]; by = yp[laneOff + 1]; bz = yp[laneOff + 2];
        }
#pragma unroll 2
        for (int t = t0; t < tMainEnd; ++t) {
            const float cx = bx, cy = by, cz = bz;
            {   // prefetch next tile (uniform pointer; overlaps the math)
                const int tn = imin(t + 1, tMainEnd - 1);
                const float* __restrict__ yp = Y + (size_t)tn * 48;
                bx = yp[laneOff + 0]; by = yp[laneOff + 1]; bz = yp[laneOff + 2];
            }
            const float b2 = cx * cx + cy * cy + cz * cz;   // lane-constant
            v2f bFrag;
            bFrag[0] = hl ? (-2.0f * cz) : (-2.0f * cx);    // K = 0 or 2
            bFrag[1] = hl ? b2           : (-2.0f * cy);    // K = 1 or 3

#pragma unroll
            for (int rt = 0; rt < RT; ++rt) {
                v8f c = {0.f, 0.f, 0.f, 0.f, 0.f, 0.f, 0.f, 0.f};
#if defined(__HIP_DEVICE_COMPILE__)
                c = __builtin_amdgcn_wmma_f32_16x16x4_f32(
                        false, aF[rt], false, bFrag, (short)0, c, false, false);
#endif
                // D layout: VGPR j -> row (rowBase + rt*16 + j + 8*hl)
#pragma unroll
                for (int j = 0; j < 8; ++j)
                    rmin[rt][j] = fmin_raw(rmin[rt][j], c[j]);
            }
        }
    }

    // ---- tail: at most one partial tile (nCols % 16 != 0); clamped lanes ---
    if (t1 > mainTiles && t0 <= mainTiles) {            // wave-uniform branch
        const int n = imin(mainTiles * 16 + idx, nCols - 1);  // dup cols OK
        const float cx = Y[n * 3 + 0], cy = Y[n * 3 + 1], cz = Y[n * 3 + 2];
        const float b2 = cx * cx + cy * cy + cz * cz;
        v2f bFrag;
        bFrag[0] = hl ? (-2.0f * cz) : (-2.0f * cx);
        bFrag[1] = hl ? b2           : (-2.0f * cy);
#pragma unroll
        for (int rt = 0; rt < RT; ++rt) {
            v8f c = {0.f, 0.f, 0.f, 0.f, 0.f, 0.f, 0.f, 0.f};
#if defined(__HIP_DEVICE_COMPILE__)
            c = __builtin_amdgcn_wmma_f32_16x16x4_f32(
                    false, aF[rt], false, bFrag, (short)0, c, false, false);
#endif
#pragma unroll
            for (int j = 0; j < 8; ++j)
                rmin[rt][j] = fmin_raw(rmin[rt][j], c[j]);
        }
    }

    // ---- epilogue: cross-lane min over 16 lanes per row, add |a|^2, clamp --
#pragma unroll
    for (int rt = 0; rt < RT; ++rt) {
#pragma unroll
        for (int j = 0; j < 8; ++j) {
            float v = rmin[rt][j];
            v = fmin_raw(v, __shfl_xor(v, 1, WAVE));
            v = fmin_raw(v, __shfl_xor(v, 2, WAVE));
            v = fmin_raw(v, __shfl_xor(v, 4, WAVE));
            v = fmin_raw(v, __shfl_xor(v, 8, WAVE));
            // |a|^2 of row (rowBase + rt*16 + 8*hl + j) lives in lane 8*hl+j
            const float a2j = __shfl(a2[rt], 8 * hl + j, WAVE);
            rmin[rt][j] = fmaxf(v + a2j, 0.0f);
        }
    }
    if (idx == 0) {
#pragma unroll
        for (int rt = 0; rt < RT; ++rt)
#pragma unroll
            for (int j = 0; j < 8; ++j) {
                const int r = imin(rowBase + rt * 16 + 8 * hl + j, nRows - 1);
                atomicMin(&rowMinBits[r], __float_as_uint(rmin[rt][j]));
            }
    }
}

// ---------------------------------------------------------------------------
// mins_seeds output: out[1+n] = sqrt(min_d2 per pred point)
// ---------------------------------------------------------------------------
__global__ void write_mins_kernel(const unsigned int* __restrict__ bitsB,
                                  float* __restrict__ out, int M) {
    int i = blockIdx.x * blockDim.x + threadIdx.x;
    if (i < M) out[1 + i] = sqrtf(__uint_as_float(bitsB[i]));
}

// ---------------------------------------------------------------------------
// Deterministic means: single block, fixed-order grid-stride + LDS tree.
// ---------------------------------------------------------------------------
__global__ void __launch_bounds__(256)
reduce_means_kernel(const unsigned int* __restrict__ bitsA,
                    const unsigned int* __restrict__ bitsB,
                    float* __restrict__ out, int N, int M) {
    __shared__ float sA[256];
    __shared__ float sB[256];
    float la = 0.f, lb = 0.f;
    for (int i = threadIdx.x; i < N; i += 256)
        la += sqrtf(__uint_as_float(bitsA[i]));
    for (int i = threadIdx.x; i < M; i += 256)
        lb += sqrtf(__uint_as_float(bitsB[i]));
    sA[threadIdx.x] = la;
    sB[threadIdx.x] = lb;
    __syncthreads();
    for (int s = 128; s > 0; s >>= 1) {
        if (threadIdx.x < (unsigned)s) {
            sA[threadIdx.x] += sA[threadIdx.x + s];
            sB[threadIdx.x] += sB[threadIdx.x + s];
        }
        __syncthreads();
    }
    if (threadIdx.x == 0) {
        float loss      = sA[0] / (float)N;  // mean over true points
        float lossSeeds = sB[0] / (float)M;  // mean over pred points
        out[0]     = loss + lossSeeds;
        out[1 + M] = loss;
        out[2 + M] = lossSeeds;
    }
}

// ---------------------------------------------------------------------------
extern "C" void kernel_launch(void* const* d_in, const int* in_sizes, int n_in,
                              void* d_out, int out_size, void* d_ws, size_t ws_size,
                              hipStream_t stream) {
    const float* truePos = (const float*)d_in[0];   // [N,3] f32
    const float* predPos = (const float*)d_in[1];   // [M,3] f32
    const int N = in_sizes[0] / 3;
    const int M = in_sizes[1] / 3;
    float* out = (float*)d_out;

    unsigned int* bitsA = (unsigned int*)d_ws;  // per-true-row min d2 bits
    unsigned int* bitsB = bitsA + N;            // per-pred-row min d2 bits

    init_bits_kernel<<<(N + M + 255) / 256, 256, 0, stream>>>(bitsA, N + M);

    const int wavesPerBlock = 8;  // 256 threads
    const int chunks = 16;        // column-tile chunks (grid.y): keep waves up

    // Pass A: rows = true points, cols = pred points -> loss
    {
        const int strips = (N + 16 * RT - 1) / (16 * RT);
        const int tiles  = (M + 15) / 16;
        dim3 grid((strips + wavesPerBlock - 1) / wavesPerBlock, chunks);
        nn_min_rows_kernel<<<grid, 256, 0, stream>>>(
            truePos, predPos, N, M, tiles, (tiles + chunks - 1) / chunks, bitsA);
    }
    // Pass B: rows = pred points, cols = true points -> loss_seeds / mins_seeds
    {
        const int strips = (M + 16 * RT - 1) / (16 * RT);
        const int tiles  = (N + 15) / 16;
        dim3 grid((strips + wavesPerBlock - 1) / wavesPerBlock, chunks);
        nn_min_rows_kernel<<<grid, 256, 0, stream>>>(
            predPos, truePos, M, N, tiles, (tiles + chunks - 1) / chunks, bitsB);
    }

    write_mins_kernel<<<(M + 255) / 256, 256, 0, stream>>>(bitsB, out, M);
    reduce_means_kernel<<<1, 256, 0, stream>>>(bitsA, bitsB, out, N, M);
}